// QuantumLSTM_58506044506792
// MI455X (gfx1250) — compile-verified
//
#include <hip/hip_runtime.h>
#include <hip/hip_bf16.h>
#include <math.h>

// ---------------- constants ----------------
#define NQ 4
#define NL 2
#define DIM 16          // 2^NQ amplitudes
#define UNITS 128
#define XD 32           // input feature dim
#define ITEMS_PER_BLOCK 128   // 4 waves x 32 items
#define INV_SQRT2 0.70710678118654752440f

typedef __attribute__((ext_vector_type(2))) float v2f;
typedef __attribute__((ext_vector_type(8))) float v8f;

// ---------------- fast branch-free activations ----------------
// tanh(x) = 1 - 2/(e^{2x}+1): e=inf -> 1, e=0 -> -1 (correct saturation)
__device__ __forceinline__ float fast_tanh(float x) {
    float e = __expf(2.f * x);
    return fmaf(-2.f, __builtin_amdgcn_rcpf(e + 1.f), 1.f);
}
__device__ __forceinline__ float fast_sigmoid(float x) {
    return __builtin_amdgcn_rcpf(1.f + __expf(-x));
}

// ---------------- complex helpers ----------------
__device__ __forceinline__ float2 cmul(float2 a, float2 b) {
    return make_float2(a.x * b.x - a.y * b.y, a.x * b.y + a.y * b.x);
}
__device__ __forceinline__ float2 cadd(float2 a, float2 b) {
    return make_float2(a.x + b.x, a.y + b.y);
}
__device__ __forceinline__ float2 cmac(float2 acc, float2 a, float2 b) {
    acc.x = fmaf(a.x, b.x, fmaf(-a.y, b.y, acc.x));
    acc.y = fmaf(a.x, b.y, fmaf(a.y, b.x, acc.y));
    return acc;
}

// apply 2x2 complex gate on wire q of a 16-amplitude state (qubit0 = MSB)
__device__ __forceinline__ void apply1(float2* st, int q,
                                       float2 u00, float2 u01,
                                       float2 u10, float2 u11) {
    int m = 8 >> q;
    #pragma unroll
    for (int k = 0; k < DIM; ++k) {
        if (!(k & m)) {
            float2 a = st[k], b = st[k | m];
            st[k]     = cadd(cmul(u00, a), cmul(u01, b));
            st[k | m] = cadd(cmul(u10, a), cmul(u11, b));
        }
    }
}
__device__ __forceinline__ void cnot(float2* st, int c, int t) {
    int mc = 8 >> c, mt = 8 >> t;
    #pragma unroll
    for (int k = 0; k < DIM; ++k) {
        if ((k & mc) && !(k & mt)) {
            float2 tmp = st[k]; st[k] = st[k | mt]; st[k | mt] = tmp;
        }
    }
}
__device__ __forceinline__ void rx_gate(float2* st, int q, float t) {
    float c = cosf(0.5f * t), s = sinf(0.5f * t);
    apply1(st, q, make_float2(c, 0.f), make_float2(0.f, -s),
                  make_float2(0.f, -s), make_float2(c, 0.f));
}
__device__ __forceinline__ void ry_gate(float2* st, int q, float t) {
    float c = cosf(0.5f * t), s = sinf(0.5f * t);
    apply1(st, q, make_float2(c, 0.f), make_float2(-s, 0.f),
                  make_float2(s, 0.f), make_float2(c, 0.f));
}
__device__ __forceinline__ void rz_gate(float2* st, int q, float t) {
    float c = cosf(0.5f * t), s = sinf(0.5f * t);
    apply1(st, q, make_float2(c, -s), make_float2(0.f, 0.f),
                  make_float2(0.f, 0.f), make_float2(c, s));
}

// ------------------------------------------------------------------
// Kernel A: build fixed 16x16 unitaries (variational layers + pre-layer
// CNOTs) by simulating all basis columns. thread = g*16 + column.
// ------------------------------------------------------------------
__global__ void qlstm_build_unitaries(const float* __restrict__ fw,
                                      const float* __restrict__ iw,
                                      const float* __restrict__ ow,
                                      const float* __restrict__ cw,
                                      float2* __restrict__ V /* [4][16][16] */) {
    int tid = threadIdx.x;
    if (tid >= 64) return;
    int g = tid >> 4, col = tid & 15;
    const float* w = (g == 0) ? fw : (g == 1) ? iw : (g == 2) ? ow : cw;

    float2 st[DIM];
    #pragma unroll
    for (int k = 0; k < DIM; ++k) st[k] = make_float2(0.f, 0.f);
    st[col] = make_float2(1.f, 0.f);

    if (g == 2) { cnot(st, 0, 1); cnot(st, 1, 2); cnot(st, 2, 3); } // output circ
    if (g == 3) { cnot(st, 0, 1); cnot(st, 2, 3); }                 // candidate circ

    for (int l = 0; l < NL; ++l) {
        for (int q = 0; q < NQ; ++q) {
            int b = (l * NQ + q) * 3;
            rx_gate(st, q, w[b]);
            ry_gate(st, q, w[b + 1]);
            rz_gate(st, q, w[b + 2]);
        }
        cnot(st, 0, 1); cnot(st, 1, 2); cnot(st, 2, 3);
    }
    #pragma unroll
    for (int r = 0; r < DIM; ++r) V[(g * DIM + r) * DIM + col] = st[r];
}

// ------------------------------------------------------------------
// Kernel B: per (b,t): angles -> kron encoding state -> fixed-unitary
// matvec -> Pauli expectations; WMMA f32 16x16x4 projection (bias in C),
// branch-free activations, store f/i/o/cand activations.
// 128 threads = 4 waves, 128 items per block.
// ------------------------------------------------------------------
__global__ __launch_bounds__(ITEMS_PER_BLOCK)
void qlstm_gates(const float* __restrict__ x,      // [items][32]
                 const float* __restrict__ Wp,     // [4][128]
                 const float* __restrict__ bp,     // [128]
                 const float2* __restrict__ Vg,    // [4][16][16]
                 float* __restrict__ actF, float* __restrict__ actI,
                 float* __restrict__ actO, float* __restrict__ actC) {
    __shared__ float2 Vl[4 * DIM * DIM];            // 8 KB
    __shared__ float  zsh[4][ITEMS_PER_BLOCK][4];   // 8 KB

    const int tid  = threadIdx.x;                   // 0..127 (one item each)
    const int lane = tid & 31;
    const int wave = tid >> 5;

    for (int i = tid; i < 4 * DIM * DIM; i += ITEMS_PER_BLOCK) Vl[i] = Vg[i];

    const int item = blockIdx.x * ITEMS_PER_BLOCK + tid;
    const float* xr = x + (size_t)item * XD;
    __builtin_prefetch(xr, 0, 0);                   // global_prefetch_b8

    // ang = l2n(x)[:4]
    float ss = 0.f;
    float x0 = 0.f, x1 = 0.f, x2 = 0.f, x3 = 0.f;
    #pragma unroll
    for (int d = 0; d < XD; d += 4) {
        float4 v = *(const float4*)(xr + d);
        ss = fmaf(v.x, v.x, ss); ss = fmaf(v.y, v.y, ss);
        ss = fmaf(v.z, v.z, ss); ss = fmaf(v.w, v.w, ss);
        if (d == 0) { x0 = v.x; x1 = v.y; x2 = v.z; x3 = v.w; }
    }
    float rs = rsqrtf(fmaxf(ss, 1e-12f));
    float ang[4] = { x0 * rs, x1 * rs, x2 * rs, x3 * rs };

    __syncthreads(); // Vl ready

    #pragma unroll
    for (int g = 0; g < 4; ++g) {
        // ---- per-qubit encoding 2-vectors ----
        float2 e0[4], e1[4];
        #pragma unroll
        for (int q = 0; q < NQ; ++q) {
            float a = ang[q];
            float ch = cosf(0.5f * a), sh = sinf(0.5f * a);
            if (g == 0) {                 // forget: RY(a)|0>
                e0[q] = make_float2(ch, 0.f);
                e1[q] = make_float2(sh, 0.f);
            } else if (g == 1) {          // input: RZ(a/2) RY(a) |0>
                float cq = cosf(0.25f * a), sq = sinf(0.25f * a);
                e0[q] = make_float2(cq * ch, -sq * ch);
                e1[q] = make_float2(cq * sh,  sq * sh);
            } else if (g == 2) {          // output: RZ(.3a) RY(.7a) RX(a) |0>
                float c7 = cosf(0.35f * a), s7 = sinf(0.35f * a);
                float2 y0 = make_float2(c7 * ch,  s7 * sh);
                float2 y1 = make_float2(s7 * ch, -c7 * sh);
                float cp = cosf(0.15f * a), sp = sinf(0.15f * a);
                e0[q] = cmul(make_float2(cp, -sp), y0);
                e1[q] = cmul(make_float2(cp,  sp), y1);
            } else {                      // candidate: RY(a) H |0>
                e0[q] = make_float2((ch - sh) * INV_SQRT2, 0.f);
                e1[q] = make_float2((ch + sh) * INV_SQRT2, 0.f);
            }
        }
        // ---- kron encoding state (qubit0 = MSB) ----
        float2 phi[DIM];
        #pragma unroll
        for (int k = 0; k < DIM; ++k) {
            float2 p = (k & 8) ? e1[0] : e0[0];
            p = cmul(p, (k & 4) ? e1[1] : e0[1]);
            p = cmul(p, (k & 2) ? e1[2] : e0[2]);
            p = cmul(p, (k & 1) ? e1[3] : e0[3]);
            phi[k] = p;
        }
        // ---- psi = V_fixed * phi ----
        float2 psi[DIM];
        #pragma unroll
        for (int r = 0; r < DIM; ++r) {
            float2 acc = make_float2(0.f, 0.f);
            const float2* row = &Vl[(g * DIM + r) * DIM];
            #pragma unroll
            for (int c = 0; c < DIM; ++c) acc = cmac(acc, row[c], phi[c]);
            psi[r] = acc;
        }
        // ---- Pauli expectations ----
        #pragma unroll
        for (int q = 0; q < NQ; ++q) {
            float z = 0.f;
            if (g == 0) {                           // <Z_q>
                int m = 8 >> q;
                #pragma unroll
                for (int k = 0; k < DIM; ++k) {
                    float p = psi[k].x * psi[k].x + psi[k].y * psi[k].y;
                    z += (k & m) ? -p : p;
                }
            } else if (g == 1) {                    // <Y_q> = 2 Sum Im(conj(a)b)
                int m = 8 >> q;
                #pragma unroll
                for (int k = 0; k < DIM; ++k) if (!(k & m)) {
                    float2 a = psi[k], b = psi[k | m];
                    z += 2.f * (a.x * b.y - a.y * b.x);
                }
            } else if (g == 2) {                    // <X_q> = 2 Sum Re(conj(a)b)
                int m = 8 >> q;
                #pragma unroll
                for (int k = 0; k < DIM; ++k) if (!(k & m)) {
                    float2 a = psi[k], b = psi[k | m];
                    z += 2.f * (a.x * b.x + a.y * b.y);
                }
            } else {                                // <Z_q X_j>, j=(q+1)%4
                int j = (q + 1) & 3;
                int mj = 8 >> j, mq = 8 >> q;
                #pragma unroll
                for (int k = 0; k < DIM; ++k) if (!(k & mj)) {
                    float2 a = psi[k], b = psi[k | mj];
                    float re = 2.f * (a.x * b.x + a.y * b.y);
                    z += (k & mq) ? -re : re;
                }
            }
            zsh[g][tid][q] = z;
        }
    }
    __syncthreads();

    // ---- projection: z[16x4] @ W_proj[4x128] + b_proj via WMMA f32 16x16x4 ----
    // Each wave projects its own 32 items (two 16-row A tiles).
    const int rowSel = lane >> 4;           // A/B: lanes 16-31 carry K=2,3
    const int kb = rowSel << 1;             // K base: 0 or 2
    const int nlane = lane & 15;
    const int itemBase = wave * 32;

    // hoist the 8 A fragments (4 gates x 2 halves) out of the unit loop
    v2f Afrag[4][2];
    #pragma unroll
    for (int g = 0; g < 4; ++g)
        #pragma unroll
        for (int h = 0; h < 2; ++h) {
            int m = itemBase + nlane + h * 16;
            Afrag[g][h].x = zsh[g][m][kb];
            Afrag[g][h].y = zsh[g][m][kb + 1];
        }

    float* dsts[4] = { actF, actI, actO, actC };

    for (int n = 0; n < 8; ++n) {           // 8 chunks of 16 units
        int colN = nlane + n * 16;
        v2f Bf;
        Bf.x = Wp[kb * UNITS + colN];
        Bf.y = Wp[(kb + 1) * UNITS + colN];
        float bb = bp[colN];
        v8f C;
        #pragma unroll
        for (int r = 0; r < 8; ++r) C[r] = bb;

        #pragma unroll
        for (int g = 0; g < 4; ++g) {
            float* dst = dsts[g];
            #pragma unroll
            for (int h = 0; h < 2; ++h) {
                v8f D = __builtin_amdgcn_wmma_f32_16x16x4_f32(
                    false, Afrag[g][h], false, Bf, (short)0, C, false, false);
                #pragma unroll
                for (int r = 0; r < 8; ++r) {
                    int mm = rowSel * 8 + r;           // output row M
                    int it = blockIdx.x * ITEMS_PER_BLOCK + itemBase + mm + h * 16;
                    float t = fast_tanh(D[r]);
                    float outv = (g == 3) ? fast_tanh(t) : fast_sigmoid(t);
                    dst[(size_t)it * UNITS + colN] = outv;
                }
            }
        }
    }
}

// ------------------------------------------------------------------
// Kernel C: elementwise LSTM-state scan over T, parallel over (b,unit).
// ------------------------------------------------------------------
__global__ void qlstm_scan(const float* __restrict__ actF,
                           const float* __restrict__ actI,
                           const float* __restrict__ actO,
                           const float* __restrict__ actC,
                           float* __restrict__ out, int T) {
    int idx = blockIdx.x * blockDim.x + threadIdx.x;   // b*UNITS + u
    int b = idx >> 7, u = idx & (UNITS - 1);
    float c = 0.f;
    size_t base = (size_t)b * T * UNITS + u;
    for (int t = 0; t < T; ++t) {
        size_t off = base + (size_t)t * UNITS;
        if (t + 1 < T) {                               // prefetch next step
            __builtin_prefetch(actF + off + UNITS, 0, 0);
            __builtin_prefetch(actI + off + UNITS, 0, 0);
            __builtin_prefetch(actO + off + UNITS, 0, 0);
            __builtin_prefetch(actC + off + UNITS, 0, 0);
        }
        float f  = actF[off];
        float i  = actI[off];
        float o  = actO[off];
        float cd = actC[off];
        c = fmaf(f, c, i * cd);
        out[off] = o * fast_tanh(c);
    }
}

// ------------------------------------------------------------------
extern "C" void kernel_launch(void* const* d_in, const int* in_sizes, int n_in,
                              void* d_out, int out_size, void* d_ws, size_t ws_size,
                              hipStream_t stream) {
    const float* x  = (const float*)d_in[0];   // [B,T,32]
    const float* fw = (const float*)d_in[1];
    const float* iw = (const float*)d_in[2];
    const float* ow = (const float*)d_in[3];
    const float* cw = (const float*)d_in[4];
    const float* Wp = (const float*)d_in[5];   // [4,128]
    const float* bp = (const float*)d_in[6];   // [128]
    float* out = (float*)d_out;

    const int items = in_sizes[0] / XD;        // B*T = 32768
    const int T = 128;
    const int B = items / T;

    const size_t NACT = (size_t)items * UNITS;
    float* ws = (float*)d_ws;
    float* actF = ws;
    float* actI = ws + NACT;
    float* actO = ws + 2 * NACT;
    float* actC = ws + 3 * NACT;
    float2* V  = (float2*)(ws + 4 * NACT);     // 4*16*16 complex

    qlstm_build_unitaries<<<1, 64, 0, stream>>>(fw, iw, ow, cw, V);

    qlstm_gates<<<items / ITEMS_PER_BLOCK, ITEMS_PER_BLOCK, 0, stream>>>(
        x, Wp, bp, V, actF, actI, actO, actC);

    qlstm_scan<<<(B * UNITS) / 128, 128, 0, stream>>>(actF, actI, actO, actC, out, T);
}